// indi_GraphSAGE_1623497638159
// MI455X (gfx1250) — compile-verified
//
#include <hip/hip_runtime.h>

// GraphSAGE forward for MI455X (gfx1250, wave32).
// fp32 WMMA (V_WMMA_F32_16X16X4_F32) for all GEMMs; L2-resident edge
// aggregation via global_atomic_add_f32.

typedef __attribute__((ext_vector_type(2))) float v2f;
typedef __attribute__((ext_vector_type(8))) float v8f;

#define HIDC 128

// ---------------------------------------------------------------- utilities
__global__ void fill_zero_kernel(float* __restrict__ p, long long n) {
  long long i = (long long)blockIdx.x * blockDim.x + threadIdx.x;
  if (i < n) p[i] = 0.0f;
}

__global__ void degree_kernel(const int* __restrict__ dst, float* __restrict__ deg, int E) {
  int e = blockIdx.x * blockDim.x + threadIdx.x;
  if (e < E) atomicAdd(&deg[dst[e]], 1.0f);
}

__global__ void invdeg_kernel(float* __restrict__ deg, int N) {
  int i = blockIdx.x * blockDim.x + threadIdx.x;
  if (i < N) deg[i] = 1.0f / fmaxf(deg[i], 1.0f);
}

// One edge handled by 32 threads; each thread moves 4 consecutive features
// (float4 gather from h[src], 4 float atomic adds into agg[dst]).
__global__ void scatter_add_kernel(const float* __restrict__ h, const int* __restrict__ src,
                                   const int* __restrict__ dst, float* __restrict__ agg, int E) {
  long long idx = (long long)blockIdx.x * blockDim.x + threadIdx.x;
  long long e = idx >> 5;
  int q = ((int)(idx & 31)) << 2;
  if (e < E) {
    int s = src[e], d = dst[e];
    const float4 v = *(const float4*)(h + (size_t)s * HIDC + q);
    float* o = agg + (size_t)d * HIDC + q;
    atomicAdd(o + 0, v.x);
    atomicAdd(o + 1, v.y);
    atomicAdd(o + 2, v.z);
    atomicAdd(o + 3, v.w);
  }
}

// ---------------------------------------------------------------- WMMA GEMM
// Cout[M x OUT] = (HAS_SCALE ? diag(rowScale) : I) * A[M x K] @ W[OUT x K]^T
//                 + bias(optional, per-col) + Cin(optional, accumulate).
//
// Block = 128 threads = 4 waves. Each wave computes one 16x16 tile:
// rows m0..m0+15 (shared by all 4 waves -> A served from WGP$ after wave 0),
// cols o0 = blockIdx.y*64 + wave*16. K consumed 4 at a time with
// V_WMMA_F32_16X16X4_F32.
//
// Out-of-range rows/cols are CLAMPED for the loads (branchless hot loop,
// EXEC stays all-ones for WMMA); the resulting garbage rows/cols are masked
// at the store. fp32 A 16x4 VGPR layout (ISA 7.12.2): lanes 0-15 row M=lane,
// VGPR0=K0, VGPR1=K1; lanes 16-31 row M=lane-16, VGPR0=K2, VGPR1=K3. B
// symmetric; C/D: VGPR r -> row r (lanes 0-15) / row r+8 (lanes 16-31),
// col = lane&15.
template <bool HAS_SCALE>
__global__ void gemm_wmma_f32_kernel(const float* __restrict__ A, const float* __restrict__ W,
                                     const float* __restrict__ bias, const float* Cin,
                                     const float* __restrict__ rowScale, float* Cout,
                                     int M, int K, int OUT, int ldc) {
  const int lane = threadIdx.x & 31;
  const int wave = threadIdx.x >> 5;
  const int half = lane >> 4;
  const int l16  = lane & 15;
  const int m0 = blockIdx.x << 4;
  const int o0 = (blockIdx.y << 6) + (wave << 4);

  int arow = m0 + l16;
  if (arow >= M) arow = M - 1;           // clamp; stores masked below
  int bcol = o0 + l16;
  if (bcol >= OUT) bcol = OUT - 1;       // clamp; garbage cols masked at store

  const float* Arow = A + (size_t)arow * K;
  const float* Wrow = W + (size_t)bcol * K;
  float rs = 1.0f;
  if (HAS_SCALE) rs = rowScale[arow];

  v8f c = {};
#pragma unroll 8
  for (int kk = 0; kk < K; kk += 4) {
    const int ka = kk + (half << 1);     // half 0 -> K{0,1}, half 1 -> K{2,3}
    const float2 av = *(const float2*)(Arow + ka);
    const float2 bv = *(const float2*)(Wrow + ka);
    v2f a;
    a[0] = av.x;
    a[1] = av.y;
    if (HAS_SCALE) { a[0] *= rs; a[1] *= rs; }
    v2f b;
    b[0] = bv.x;
    b[1] = bv.y;
    // 8 args: (neg_a, A, neg_b, B, c_mod, C, reuse_a, reuse_b)
    c = __builtin_amdgcn_wmma_f32_16x16x4_f32(false, a, false, b, (short)0, c, false, false);
  }

  const int n = o0 + l16;
  if (n < OUT) {
    const float badd = (bias != nullptr) ? bias[n] : 0.0f;
#pragma unroll
    for (int r = 0; r < 8; ++r) {
      const int m = m0 + (half << 3) + r;
      if (m < M) {
        const size_t off = (size_t)m * ldc + n;
        float v = c[r] + badd;
        if (Cin) v += Cin[off];
        Cout[off] = v;
      }
    }
  }
}

// ---------------------------------------------------------------- BatchNorm
__global__ void bn_stats_kernel(const float* __restrict__ h, float* __restrict__ stats, int N) {
  const int f = threadIdx.x & (HIDC - 1);
  const int rpb = blockDim.x / HIDC;            // rows handled per block per pass
  int r = blockIdx.x * rpb + (threadIdx.x / HIDC);
  const int stride = gridDim.x * rpb;
  float s = 0.0f, sq = 0.0f;
  for (; r < N; r += stride) {
    const float v = h[(size_t)r * HIDC + f];
    s += v;
    sq += v * v;
  }
  atomicAdd(&stats[f], s);
  atomicAdd(&stats[HIDC + f], sq);
}

__global__ void bn_finalize_kernel(float* __restrict__ stats, const float* __restrict__ scale,
                                   const float* __restrict__ bias, int N, float eps) {
  const int f = threadIdx.x;
  if (f < HIDC) {
    const float invN = 1.0f / (float)N;
    const float mean = stats[f] * invN;
    float var = stats[HIDC + f] * invN - mean * mean;   // biased variance
    var = fmaxf(var, 0.0f);
    const float a = scale[f] * rsqrtf(var + eps);
    const float b = bias[f] - mean * a;
    stats[f] = a;
    stats[HIDC + f] = b;
  }
}

__global__ void bn_apply_relu_kernel(float* __restrict__ h, const float* __restrict__ stats,
                                     long long total) {
  long long i = (long long)blockIdx.x * blockDim.x + threadIdx.x;
  if (i < total) {
    const int f = (int)(i & (HIDC - 1));
    const float v = h[i] * stats[f] + stats[HIDC + f];
    h[i] = fmaxf(v, 0.0f);
  }
}

// ---------------------------------------------------------------- launcher
extern "C" void kernel_launch(void* const* d_in, const int* in_sizes, int n_in,
                              void* d_out, int out_size, void* d_ws, size_t ws_size,
                              hipStream_t stream) {
  const float* x     = (const float*)d_in[0];
  const int*   ei    = (const int*)d_in[1];
  const float* fc0_w = (const float*)d_in[2];
  const float* fc0_b = (const float*)d_in[3];
  const float* bnsc  = (const float*)d_in[4];
  const float* bnbi  = (const float*)d_in[5];
  const float* lin_l = (const float*)d_in[6];
  const float* lin_r = (const float*)d_in[7];
  const float* W_w   = (const float*)d_in[8];
  const float* W_b   = (const float*)d_in[9];
  const float* out_w = (const float*)d_in[10];
  const float* out_b = (const float*)d_in[11];
  float* out = (float*)d_out;

  const int N    = in_sizes[0] / HIDC;   // 50000
  const int E    = in_sizes[1] / 2;      // 640000
  const int NCLS = in_sizes[11];         // 40
  const float eps = 1e-5f;

  const int* src = ei;                   // edge_index[0]
  const int* dst = ei + E;               // edge_index[1]

  // workspace layout (floats): h | h2 | agg | inv_deg | stats
  float* h      = (float*)d_ws;
  float* h2     = h   + (size_t)N * HIDC;
  float* agg    = h2  + (size_t)N * HIDC;
  float* invdeg = agg + (size_t)N * HIDC;
  float* stats  = invdeg + N;

  const dim3 B128(128), B256(256);
  const int MT = (N + 15) / 16;

  auto gemm = [&](const float* A, const float* Wm, const float* bias, const float* Cin,
                  const float* rowScale, float* Cout, int OUT, int ldc) {
    dim3 grid((unsigned)MT, (unsigned)((OUT + 63) / 64));
    if (rowScale)
      gemm_wmma_f32_kernel<true><<<grid, B128, 0, stream>>>(A, Wm, bias, Cin, rowScale, Cout,
                                                            N, HIDC, OUT, ldc);
    else
      gemm_wmma_f32_kernel<false><<<grid, B128, 0, stream>>>(A, Wm, bias, Cin, nullptr, Cout,
                                                             N, HIDC, OUT, ldc);
  };
  auto fillz = [&](float* p, long long n) {
    fill_zero_kernel<<<dim3((unsigned)((n + 255) / 256)), B256, 0, stream>>>(p, n);
  };
  auto bn = [&](float* hb, int layer) {
    fillz(stats, 2 * HIDC);
    bn_stats_kernel<<<dim3(256), B256, 0, stream>>>(hb, stats, N);
    bn_finalize_kernel<<<dim3(1), B128, 0, stream>>>(stats, bnsc + (size_t)layer * HIDC,
                                                     bnbi + (size_t)layer * HIDC, N, eps);
    bn_apply_relu_kernel<<<dim3((unsigned)(((long long)N * HIDC + 255) / 256)), B256, 0,
                           stream>>>(hb, stats, (long long)N * HIDC);
  };

  // --- degrees -> inv_deg
  fillz(invdeg, N);
  degree_kernel<<<dim3((unsigned)((E + 255) / 256)), B256, 0, stream>>>(dst, invdeg, E);
  invdeg_kernel<<<dim3((unsigned)((N + 255) / 256)), B256, 0, stream>>>(invdeg, N);

  // --- input linear + BN + ReLU
  gemm(x, fc0_w, fc0_b, nullptr, nullptr, h, HIDC, HIDC);
  bn(h, 0);

  // --- 2 SAGE layers
  for (int layer = 0; layer < 2; ++layer) {
    fillz(agg, (long long)N * HIDC);
    const long long sthreads = (long long)E * 32;
    scatter_add_kernel<<<dim3((unsigned)((sthreads + 255) / 256)), B256, 0, stream>>>(
        h, src, dst, agg, E);
    // h2 = (inv_deg * agg) @ lin_l^T          (mean-agg fused into A-load)
    gemm(agg, lin_l + (size_t)layer * HIDC * HIDC, nullptr, nullptr, invdeg, h2, HIDC, HIDC);
    // h2 += h @ lin_r^T
    gemm(h, lin_r + (size_t)layer * HIDC * HIDC, nullptr, h2, nullptr, h2, HIDC, HIDC);
    // h = h2 @ W_w^T + W_b
    gemm(h2, W_w, W_b, nullptr, nullptr, h, HIDC, HIDC);
    bn(h, layer + 1);
  }

  // --- classifier head (OUT=40, column-masked tiles)
  gemm(h, out_w, out_b, nullptr, nullptr, out, NCLS, NCLS);
}